// GNN_3582002725394
// MI455X (gfx1250) — compile-verified
//
#include <hip/hip_runtime.h>

// ---------------------------------------------------------------------------
// GINE-style GNN, 4 layers, N=50000 nodes, E=500000 edges, D=128 features.
// FP32 throughout (matches reference precision; compute is free — edge-phase
// HBM traffic ~1 GB/layer dominates at 23.3 TB/s). GEMM runs on the FP32 WMMA
// pipe: V_WMMA_F32_16X16X4_F32, with W staged in LDS in a row-pair-interleaved
// layout so each B fragment is one ds_load_b64 (no VALU repacking).
// ---------------------------------------------------------------------------

static constexpr int FD = 128;          // feature dim
static constexpr float BN_EPS = 1e-5f;

typedef float v2f __attribute__((ext_vector_type(2)));
typedef float v8f __attribute__((ext_vector_type(8)));

// ---------------- zero agg + BN stats ----------------
__global__ __launch_bounds__(256) void gine_zero(float* __restrict__ agg,
                                                 float* __restrict__ stats,
                                                 int n4) {
    int tid = blockIdx.x * 256 + threadIdx.x;
    float4 z = make_float4(0.f, 0.f, 0.f, 0.f);
    if (tid < n4) ((float4*)agg)[tid] = z;
    if (tid < (2 * FD) / 4) ((float4*)stats)[tid] = z;
}

// ---------------- edge scatter: agg[dst] += relu(x[src] + e) ----------------
// One wave per edge; lane l owns features [4l, 4l+4) -> 512B coalesced per wave.
__global__ __launch_bounds__(256) void gine_scatter(const float* __restrict__ x,
                                                    const int* __restrict__ ei,
                                                    const float* __restrict__ ea,
                                                    float* __restrict__ agg,
                                                    int E) {
    int e = blockIdx.x * 8 + (threadIdx.x >> 5);
    if (e >= E) return;
    int lane = threadIdx.x & 31;
    int c = lane * 4;
    int src = ei[e];
    int dst = ei[E + e];
    float4 xv = *(const float4*)(x + (size_t)src * FD + c);
    float4 ev = *(const float4*)(ea + (size_t)e * FD + c);
    float4 m;
    m.x = fmaxf(xv.x + ev.x, 0.f);
    m.y = fmaxf(xv.y + ev.y, 0.f);
    m.z = fmaxf(xv.z + ev.z, 0.f);
    m.w = fmaxf(xv.w + ev.w, 0.f);
    float* p = agg + (size_t)dst * FD + c;
    atomicAdd(p + 0, m.x);
    atomicAdd(p + 1, m.y);
    atomicAdd(p + 2, m.z);
    atomicAdd(p + 3, m.w);
}

// ---------------- h = relu((x + agg) @ W), in place over agg ----------------
// Block: 256 thr (8 waves) -> 64 rows x 128 cols. A tile fully staged in LDS
// before any store, so in-place update of agg is race-free within the block
// and rows are block-exclusive across the grid.
//
// Bs layout (row-pair interleave): element W[ks+2p+h][n] lives at
//   Bs[p*256 + n*2 + h],  p in [0,32), h in {0,1}, n in [0,128)
// so a B fragment for K rows {k, k+1} at column n is one aligned v2f load.
__global__ __launch_bounds__(256) void gine_gemm_relu(const float* __restrict__ xin,
                                                      float* __restrict__ h,  // in: agg, out: relu((x+agg)W)
                                                      const float* __restrict__ W,
                                                      int n) {
    __shared__ float As[64 * FD];   // 32 KB: A = x + agg, rows row0..row0+63
    __shared__ float Bs[64 * FD];   // 32 KB: 64-row chunk of W, pair-interleaved
    const int tid  = threadIdx.x;
    const int row0 = blockIdx.x * 64;

    // Stage A = x + agg (zero-pad rows >= n)
#pragma unroll
    for (int i = 0; i < 8; ++i) {
        int lin = (i * 256 + tid) * 4;
        int r = lin >> 7, c = lin & 127;
        int gr = row0 + r;
        float4 v = make_float4(0.f, 0.f, 0.f, 0.f);
        if (gr < n) {
            float4 xv = *(const float4*)(xin + (size_t)gr * FD + c);
            float4 av = *(const float4*)(h + (size_t)gr * FD + c);
            v.x = xv.x + av.x; v.y = xv.y + av.y;
            v.z = xv.z + av.z; v.w = xv.w + av.w;
        }
        *(float4*)(As + lin) = v;
    }

    const int wave   = tid >> 5;
    const int lane   = tid & 31;
    const int rt     = wave >> 1;         // row tile 0..3 (16 rows each)
    const int ctbase = (wave & 1) * 64;   // 4 col tiles of 16 per wave
    const int lhalf  = lane >> 4;         // lane half selects K pair
    const int l16    = lane & 15;

    v8f acc[4] = {};

    for (int ks = 0; ks < FD; ks += 64) {
        __syncthreads();                  // covers A staging / previous Bs use
        // Stage W rows ks..ks+63 pair-interleaved: unit u -> rows 2p,2p+1,
        // cols {cp, cp+1}; two 8B coalesced global loads, one ds_store_b128.
#pragma unroll
        for (int i = 0; i < 8; ++i) {
            int u  = i * 256 + tid;       // 0..2047
            int p  = u >> 6;              // row pair 0..31
            int cp = (u & 63) * 2;        // column pair base 0..126
            v2f r0 = *(const v2f*)(W + (size_t)(ks + 2 * p + 0) * FD + cp);
            v2f r1 = *(const v2f*)(W + (size_t)(ks + 2 * p + 1) * FD + cp);
            float4 pk;
            pk.x = r0.x; pk.y = r1.x;     // col cp:   rows 2p, 2p+1
            pk.z = r0.y; pk.w = r1.y;     // col cp+1: rows 2p, 2p+1
            *(float4*)(Bs + p * 256 + cp * 2) = pk;
        }
        __syncthreads();
#pragma unroll
        for (int k0 = 0; k0 < 64; k0 += 4) {
            const int ka = k0 + lhalf * 2;
            const int p  = (k0 >> 1) + lhalf;   // K row-pair index in Bs
            // A frag 16x4 f32: lanes 0-15 K={k0,k0+1}, lanes 16-31 K={k0+2,k0+3}
            v2f a = *(const v2f*)(As + (rt * 16 + l16) * FD + ks + ka);
#pragma unroll
            for (int t = 0; t < 4; ++t) {
                const int nb = ctbase + t * 16 + l16;
                // B frag 4x16 f32: rows {ka, ka+1} at column nb -> one b64 load
                v2f b = *(const v2f*)(Bs + p * 256 + nb * 2);
                acc[t] = __builtin_amdgcn_wmma_f32_16x16x4_f32(
                    /*neg_a=*/false, a, /*neg_b=*/false, b,
                    /*c_mod=*/(short)0, acc[t],
                    /*reuse_a=*/false, /*reuse_b=*/false);
            }
        }
    }

    // C/D layout: VGPR v -> lanes 0-15: M=v, lanes 16-31: M=v+8; N = lane&15
#pragma unroll
    for (int t = 0; t < 4; ++t) {
#pragma unroll
        for (int v = 0; v < 8; ++v) {
            int gr = row0 + rt * 16 + lhalf * 8 + v;
            if (gr < n)
                h[(size_t)gr * FD + ctbase + t * 16 + l16] = fmaxf(acc[t][v], 0.f);
        }
    }
}

// ---------------- BN stats: per-feature sum / sumsq ----------------
__global__ __launch_bounds__(128) void gine_bn_stats(const float* __restrict__ h,
                                                     float* __restrict__ stats,
                                                     int n) {
    int col = threadIdx.x;                // 128 features
    int r0 = blockIdx.x * 256;
    int rend = min(r0 + 256, n);
    float s = 0.f, s2 = 0.f;
    for (int r = r0; r < rend; ++r) {
        float v = h[(size_t)r * FD + col];
        s += v;
        s2 += v * v;
    }
    atomicAdd(&stats[col], s);
    atomicAdd(&stats[FD + col], s2);
}

// ---------------- xout = (h - mu)*rsqrt(var+eps)*g + b + xin ----------------
__global__ __launch_bounds__(256) void gine_bn_residual(const float* __restrict__ h,
                                                        const float* __restrict__ xin,
                                                        const float* __restrict__ stats,
                                                        const float* __restrict__ g,
                                                        const float* __restrict__ b,
                                                        float* __restrict__ xout,
                                                        int n) {
    int tid = blockIdx.x * 256 + threadIdx.x;
    if (tid >= n * (FD / 4)) return;
    int row = tid >> 5;
    int c = (tid & 31) * 4;
    float invN = 1.f / (float)n;
    float4 hv = *(const float4*)(h + (size_t)row * FD + c);
    float4 xv = *(const float4*)(xin + (size_t)row * FD + c);
    const float* hp = (const float*)&hv;
    const float* xp = (const float*)&xv;
    float4 o;
    float* op = (float*)&o;
#pragma unroll
    for (int j = 0; j < 4; ++j) {
        float mu  = stats[c + j] * invN;
        float var = stats[FD + c + j] * invN - mu * mu;
        float rs  = rsqrtf(var + BN_EPS);
        op[j] = (hp[j] - mu) * rs * g[c + j] + b[c + j] + xp[j];
    }
    *(float4*)(xout + (size_t)row * FD + c) = o;
}

// ---------------------------------------------------------------------------
extern "C" void kernel_launch(void* const* d_in, const int* in_sizes, int n_in,
                              void* d_out, int out_size, void* d_ws, size_t ws_size,
                              hipStream_t stream) {
    const float* x     = (const float*)d_in[0];
    const int*   ei    = (const int*)d_in[1];   // [2, E]: src then dst
    const float* ea    = (const float*)d_in[2];
    /* d_in[3] = batch: unused (single graph) */
    const float* Ws    = (const float*)d_in[4]; // [L, D, D]
    const float* gamma = (const float*)d_in[5]; // [L, D]
    const float* beta  = (const float*)d_in[6]; // [L, D]
    float* out = (float*)d_out;

    const int n = in_sizes[0] / FD;             // 50000
    const int e = in_sizes[1] / 2;              // 500000
    const int L = in_sizes[4] / (FD * FD);      // 4

    float* ws = (float*)d_ws;
    const size_t nd = (size_t)n * FD;
    float* bufA  = ws;
    float* bufB  = ws + nd;
    float* agg   = ws + 2 * nd;                 // also holds h after GEMM
    float* stats = ws + 3 * nd;                 // 2*FD floats

    const int n4 = (int)(nd / 4);
    const float* xin = x;
    for (int l = 0; l < L; ++l) {
        float* xout = (l == L - 1) ? out : ((l & 1) ? bufB : bufA);

        gine_zero<<<(n4 + 255) / 256, 256, 0, stream>>>(agg, stats, n4);
        gine_scatter<<<(e + 7) / 8, 256, 0, stream>>>(xin, ei, ea, agg, e);
        gine_gemm_relu<<<(n + 63) / 64, 256, 0, stream>>>(
            xin, agg, Ws + (size_t)l * FD * FD, n);
        gine_bn_stats<<<(n + 255) / 256, 128, 0, stream>>>(agg, stats, n);
        gine_bn_residual<<<(n * (FD / 4) + 255) / 256, 256, 0, stream>>>(
            agg, xin, stats, gamma + (size_t)l * FD, beta + (size_t)l * FD, xout, n);

        xin = xout;
    }
}